// TorchFizzBuzz_65773129171765
// MI455X (gfx1250) — compile-verified
//
#include <hip/hip_runtime.h>
#include <hip/hip_bf16.h>

// FizzBuzz class counts over [0, n) in closed form:
//   fizzbuzz = #mult6        = (n+5)/6
//   buzz     = #mult3-#mult6 = (n+2)/3 - (n+5)/6
//   fizz     = #even -#mult6 = (n+1)/2 - (n+5)/6
// Output order per reference: stack([fizz, buzz, fizzbuzz]) -> out[0..2].
//
// Launch-latency bound on MI455X: 4 bytes read, a handful of SALU integer
// ops (uniform n -> magic-number division in scalar unit), 12 bytes written
// by one predicated, coalesced global_store_b32 (lanes 0..2 of one wave32).
__global__ void TorchFizzBuzz_65773129171765_kernel(const int* __restrict__ n_ptr,
                                                    float* __restrict__ out) {
    const int n  = n_ptr[0];        // uniform -> s_load_b32
    const int m6 = (n + 5) / 6;     // multiples of 6 in [0, n)
    const int m3 = (n + 2) / 3;     // multiples of 3 in [0, n)
    const int m2 = (n + 1) / 2;     // even numbers in [0, n)

    const float fizz     = (float)(m2 - m6);
    const float buzz     = (float)(m3 - m6);
    const float fizzbuzz = (float)m6;

    // Branchless lane select + single predicated store (lanes 0..2).
    const unsigned lane = threadIdx.x;
    float v = fizz;
    v = (lane == 1u) ? buzz     : v;
    v = (lane == 2u) ? fizzbuzz : v;
    if (lane < 3u) {
        out[lane] = v;
    }
}

extern "C" void kernel_launch(void* const* d_in, const int* in_sizes, int n_in,
                              void* d_out, int out_size, void* d_ws, size_t ws_size,
                              hipStream_t stream) {
    (void)in_sizes; (void)n_in; (void)out_size; (void)d_ws; (void)ws_size;
    const int* n_ptr = (const int*)d_in[0];
    float* out = (float*)d_out;
    // One wave32 block: minimal launch footprint, deterministic, capture-safe.
    TorchFizzBuzz_65773129171765_kernel<<<1, 32, 0, stream>>>(n_ptr, out);
}